// SAGE_79328045957727
// MI455X (gfx1250) — compile-verified
//
#include <hip/hip_runtime.h>
#include <hip/hip_bf16.h>

// ---------------------------------------------------------------------------
// GraphSAGE (3x SAGEConv + shared BN + decoder MLP) for MI455X / gfx1250.
// fp32 end-to-end using V_WMMA_F32_16X16X4_F32 (wave32).
// A-tiles staged to LDS via GLOBAL_LOAD_ASYNC_TO_LDS_B128 when available.
// ---------------------------------------------------------------------------

#define N_NODES 100000
#define N_EDGES 1600000
#define HID     128
#define APITCH  132   // LDS row pitch for A tiles (bank-conflict-free b64 frags)

typedef __attribute__((ext_vector_type(2))) float v2f;
typedef __attribute__((ext_vector_type(8))) float v8f;
typedef __attribute__((ext_vector_type(4))) int   v4i;

#if __has_builtin(__builtin_amdgcn_global_load_async_to_lds_b128) && \
    __has_builtin(__builtin_amdgcn_s_wait_asynccnt)
#define HAVE_ASYNC_LDS 1
typedef __attribute__((address_space(1))) v4i* gas_v4i_ptr;  // global int4*
typedef __attribute__((address_space(3))) v4i* las_v4i_ptr;  // LDS int4*
#else
#define HAVE_ASYNC_LDS 0
#endif

// ---------------------------------------------------------------- fill zeros
__global__ void k_fill0(float* __restrict__ p, size_t n) {
    size_t i = (size_t)blockIdx.x * blockDim.x + threadIdx.x;
    if (i < n) p[i] = 0.0f;
}

// ---------------------------------------------------------------- degrees
__global__ void k_degree(const int* __restrict__ dst, float* __restrict__ deg, int ne) {
    int e = blockIdx.x * blockDim.x + threadIdx.x;
    if (e < ne) atomicAdd(&deg[dst[e]], 1.0f);
}

// ------------------------------------------------- edge gather + segment sum
// One wave per edge: 32 lanes x float4 = 128 features. h is L2-resident
// (51.2 MB << 192 MB L2), so the random gather + f32 atomics run at L2 speed.
__global__ void k_aggregate(const float* __restrict__ h,
                            const int* __restrict__ src,
                            const int* __restrict__ dst,
                            float* __restrict__ nsum, int ne) {
    int lane = threadIdx.x & 31;
    int e = blockIdx.x * (blockDim.x >> 5) + (threadIdx.x >> 5);
    if (e >= ne) return;
    int s = src[e];
    int d = dst[e];
    float4 v = ((const float4*)(h + (size_t)s * HID))[lane];
    float* p = nsum + (size_t)d * HID + lane * 4;
    atomicAdd(p + 0, v.x);
    atomicAdd(p + 1, v.y);
    atomicAdd(p + 2, v.z);
    atomicAdd(p + 3, v.w);
}

// ------------------------------------------------------- fused WMMA GEMM
// out[r,:] = A[r,:] @ W1 (+ (Nsum[r,:]/deg[r]) @ W2) + bias   (+ReLU / +stats)
// Block: 128 threads = 4 waves. Block tile: 32 rows x 128 cols.
// Wave tile: 32 rows x 32 cols = 2x2 WMMA 16x16 accumulators, K-step 4.
template <bool DUAL, bool RELU, bool STATS>
__launch_bounds__(128)
__global__ void k_sage_gemm(const float* __restrict__ A,
                            const float* __restrict__ Nsum,
                            const float* __restrict__ deg,
                            const float* __restrict__ W1,
                            const float* __restrict__ W2,
                            const float* __restrict__ bias,
                            float* __restrict__ out,
                            float* __restrict__ colsum,
                            float* __restrict__ colsumsq) {
    __shared__ float sA1[32 * APITCH];
    __shared__ float sA2[32 * APITCH];

    const int tid  = threadIdx.x;
    const int lane = tid & 31;
    const int wc   = tid >> 5;          // wave id -> 32-col group
    const int r0   = blockIdx.x * 32;   // 3125 blocks * 32 == 100000 exactly

    // Warm L1 with the weight matrices (128 threads x 512B == 64KB each).
    __builtin_prefetch(W1 + (size_t)tid * 128, 0, 3);
    if (DUAL) __builtin_prefetch(W2 + (size_t)tid * 128, 0, 3);

    // Stage A tiles into LDS: 32 rows x 128 floats per matrix.
    for (int i = tid; i < 32 * 32; i += 128) {
        int row = i >> 5, q = i & 31;
#if HAVE_ASYNC_LDS
        // Direct memory->LDS async copy (no VGPR round trip), ASYNCcnt path.
        __builtin_amdgcn_global_load_async_to_lds_b128(
            (gas_v4i_ptr)(A + (size_t)(r0 + row) * HID + q * 4),
            (las_v4i_ptr)(&sA1[row * APITCH + q * 4]),
            0, 0);
#else
        float4 v = ((const float4*)(A + (size_t)(r0 + row) * HID))[q];
        float* d1 = &sA1[row * APITCH + q * 4];
        d1[0] = v.x; d1[1] = v.y; d1[2] = v.z; d1[3] = v.w;
#endif
        if (DUAL) {
            // neigh_mean needs the 1/deg scaling, so this half stays manual.
            float inv = 1.0f / fmaxf(deg[r0 + row], 1.0f);
            float4 u = ((const float4*)(Nsum + (size_t)(r0 + row) * HID))[q];
            float* d2 = &sA2[row * APITCH + q * 4];
            d2[0] = u.x * inv; d2[1] = u.y * inv; d2[2] = u.z * inv; d2[3] = u.w * inv;
        }
    }
#if HAVE_ASYNC_LDS
    __builtin_amdgcn_s_wait_asynccnt(0);
#endif
    __syncthreads();

    const int half = lane >> 4;   // K-half select (K=0,1 vs K=2,3)
    const int l16  = lane & 15;
    const int cb   = wc * 32;     // wave column base

    v8f acc[2][2] = {};

#pragma unroll 4
    for (int k = 0; k < HID; k += 4) {
        const int kk = k + 2 * half;
        // A fragments (LDS, conflict-free b64 thanks to pitch 132)
        v2f a1_0 = *(const v2f*)&sA1[(l16)      * APITCH + kk];
        v2f a1_1 = *(const v2f*)&sA1[(16 + l16) * APITCH + kk];
        // B fragments for W1 (global; weights are L2/L1 resident broadcasts)
        v2f b1_0, b1_1;
        b1_0.x = W1[(kk)     * HID + cb + l16];
        b1_0.y = W1[(kk + 1) * HID + cb + l16];
        b1_1.x = W1[(kk)     * HID + cb + 16 + l16];
        b1_1.y = W1[(kk + 1) * HID + cb + 16 + l16];

        acc[0][0] = __builtin_amdgcn_wmma_f32_16x16x4_f32(false, a1_0, false, b1_0,
                                                          (short)0, acc[0][0], false, false);
        acc[0][1] = __builtin_amdgcn_wmma_f32_16x16x4_f32(false, a1_0, false, b1_1,
                                                          (short)0, acc[0][1], false, false);
        acc[1][0] = __builtin_amdgcn_wmma_f32_16x16x4_f32(false, a1_1, false, b1_0,
                                                          (short)0, acc[1][0], false, false);
        acc[1][1] = __builtin_amdgcn_wmma_f32_16x16x4_f32(false, a1_1, false, b1_1,
                                                          (short)0, acc[1][1], false, false);

        if (DUAL) {
            v2f a2_0 = *(const v2f*)&sA2[(l16)      * APITCH + kk];
            v2f a2_1 = *(const v2f*)&sA2[(16 + l16) * APITCH + kk];
            v2f b2_0, b2_1;
            b2_0.x = W2[(kk)     * HID + cb + l16];
            b2_0.y = W2[(kk + 1) * HID + cb + l16];
            b2_1.x = W2[(kk)     * HID + cb + 16 + l16];
            b2_1.y = W2[(kk + 1) * HID + cb + 16 + l16];

            acc[0][0] = __builtin_amdgcn_wmma_f32_16x16x4_f32(false, a2_0, false, b2_0,
                                                              (short)0, acc[0][0], false, false);
            acc[0][1] = __builtin_amdgcn_wmma_f32_16x16x4_f32(false, a2_0, false, b2_1,
                                                              (short)0, acc[0][1], false, false);
            acc[1][0] = __builtin_amdgcn_wmma_f32_16x16x4_f32(false, a2_1, false, b2_0,
                                                              (short)0, acc[1][0], false, false);
            acc[1][1] = __builtin_amdgcn_wmma_f32_16x16x4_f32(false, a2_1, false, b2_1,
                                                              (short)0, acc[1][1], false, false);
        }
    }

    // Epilogue: bias (+ReLU), store, optional per-column BN statistics.
    const float bsum[2] = { bias[cb + l16], bias[cb + 16 + l16] };
#pragma unroll
    for (int rt = 0; rt < 2; ++rt) {
#pragma unroll
        for (int ct = 0; ct < 2; ++ct) {
            float s = 0.0f, sq = 0.0f;
#pragma unroll
            for (int i = 0; i < 8; ++i) {
                float v = acc[rt][ct][i] + bsum[ct];
                if (RELU) v = fmaxf(v, 0.0f);
                int row = r0 + rt * 16 + 8 * half + i;
                out[(size_t)row * HID + cb + ct * 16 + l16] = v;
                if (STATS) { s += v; sq += v * v; }
            }
            if (STATS) {
                s  += __shfl_xor(s, 16, 32);
                sq += __shfl_xor(sq, 16, 32);
                if (half == 0) {
                    atomicAdd(&colsum[cb + ct * 16 + l16], s);
                    atomicAdd(&colsumsq[cb + ct * 16 + l16], sq);
                }
            }
        }
    }
}

// ------------------------------------------------- BN: finalize mu/var
__global__ void k_bn_finalize(const float* __restrict__ colsum,
                              const float* __restrict__ colsumsq,
                              const float* __restrict__ gamma,
                              const float* __restrict__ beta,
                              float* __restrict__ scale,
                              float* __restrict__ shift, float invN) {
    int i = threadIdx.x;  // 128 threads
    float mu  = colsum[i] * invN;
    float var = colsumsq[i] * invN - mu * mu;   // biased var = E[x^2]-mu^2
    float sc  = gamma[i] * rsqrtf(var + 1e-5f);
    scale[i] = sc;
    shift[i] = beta[i] - mu * sc;
}

// ------------------------------------------------- BN apply + ReLU (in place)
__global__ void k_bn_relu(float* __restrict__ h,
                          const float* __restrict__ scale,
                          const float* __restrict__ shift, size_t total) {
    size_t i = (size_t)blockIdx.x * blockDim.x + threadIdx.x;
    if (i < total) {
        int c = (int)(i & (HID - 1));
        h[i] = fmaxf(fmaf(h[i], scale[c], shift[c]), 0.0f);
    }
}

// ------------------------------------------------- final 128->1 projection
__global__ void k_dot(const float* __restrict__ h, const float* __restrict__ w,
                      const float* __restrict__ b, float* __restrict__ out, int n) {
    int lane = threadIdx.x & 31;
    int row = blockIdx.x * (blockDim.x >> 5) + (threadIdx.x >> 5);
    if (row >= n) return;
    float4 hv = ((const float4*)(h + (size_t)row * HID))[lane];
    float4 wv = ((const float4*)w)[lane];
    float s = hv.x * wv.x + hv.y * wv.y + hv.z * wv.z + hv.w * wv.w;
#pragma unroll
    for (int off = 16; off > 0; off >>= 1) s += __shfl_xor(s, off, 32);
    if (lane == 0) out[row] = s + b[0];
}

// ---------------------------------------------------------------------------
extern "C" void kernel_launch(void* const* d_in, const int* in_sizes, int n_in,
                              void* d_out, int out_size, void* d_ws, size_t ws_size,
                              hipStream_t stream) {
    (void)in_sizes; (void)n_in; (void)out_size; (void)ws_size;

    const float* x      = (const float*)d_in[0];
    const int*   src    = (const int*)d_in[1];
    const int*   dst    = (const int*)d_in[2];
    const float* Wself1 = (const float*)d_in[3];
    const float* Wneigh1= (const float*)d_in[4];
    const float* b1     = (const float*)d_in[5];
    const float* Wself2 = (const float*)d_in[6];
    const float* Wneigh2= (const float*)d_in[7];
    const float* b2     = (const float*)d_in[8];
    const float* Wself3 = (const float*)d_in[9];
    const float* Wneigh3= (const float*)d_in[10];
    const float* b3     = (const float*)d_in[11];
    const float* gamma  = (const float*)d_in[12];
    const float* beta   = (const float*)d_in[13];
    const float* Wd1    = (const float*)d_in[14];
    const float* bd1    = (const float*)d_in[15];
    const float* Wd2    = (const float*)d_in[16];
    const float* bd2    = (const float*)d_in[17];
    const float* Wd3    = (const float*)d_in[18];
    const float* bd3    = (const float*)d_in[19];
    float* outp = (float*)d_out;

    // Workspace layout (floats): h0 | h1 | nsum | deg | colsum | colsumsq | scale | shift
    const size_t NH = (size_t)N_NODES * HID;
    float* ws       = (float*)d_ws;
    float* h0       = ws;
    float* h1       = ws + NH;
    float* nsum     = ws + 2 * NH;
    float* deg      = ws + 3 * NH;
    float* colsum   = deg + N_NODES;
    float* colsumsq = colsum + HID;
    float* scale    = colsumsq + HID;
    float* shift    = scale + HID;

    const int TB = 256;
    const int fillNH_g   = (int)((NH + TB - 1) / TB);
    const int fillDeg_g  = (N_NODES + TB - 1) / TB;
    const int edgeWave_g = N_EDGES / (TB / 32);           // 200000
    const int edgeThr_g  = (N_EDGES + TB - 1) / TB;       // 6250
    const int gemm_g     = N_NODES / 32;                  // 3125 (exact)
    const int bnrelu_g   = fillNH_g;
    const int dot_g      = N_NODES / (TB / 32);           // 12500 (exact)
    const float invN     = 1.0f / (float)N_NODES;

    // ---- degree (once) ----
    k_fill0<<<fillDeg_g, TB, 0, stream>>>(deg, (size_t)N_NODES);
    k_degree<<<edgeThr_g, TB, 0, stream>>>(dst, deg, N_EDGES);

    // ---- layer 1 ----
    k_fill0<<<fillNH_g, TB, 0, stream>>>(nsum, NH);
    k_fill0<<<1, TB, 0, stream>>>(colsum, (size_t)2 * HID);
    k_aggregate<<<edgeWave_g, TB, 0, stream>>>(x, src, dst, nsum, N_EDGES);
    k_sage_gemm<true, false, true><<<gemm_g, 128, 0, stream>>>(
        x, nsum, deg, Wself1, Wneigh1, b1, h0, colsum, colsumsq);
    k_bn_finalize<<<1, HID, 0, stream>>>(colsum, colsumsq, gamma, beta, scale, shift, invN);
    k_bn_relu<<<bnrelu_g, TB, 0, stream>>>(h0, scale, shift, NH);

    // ---- layer 2 ----
    k_fill0<<<fillNH_g, TB, 0, stream>>>(nsum, NH);
    k_fill0<<<1, TB, 0, stream>>>(colsum, (size_t)2 * HID);
    k_aggregate<<<edgeWave_g, TB, 0, stream>>>(h0, src, dst, nsum, N_EDGES);
    k_sage_gemm<true, false, true><<<gemm_g, 128, 0, stream>>>(
        h0, nsum, deg, Wself2, Wneigh2, b2, h1, colsum, colsumsq);
    k_bn_finalize<<<1, HID, 0, stream>>>(colsum, colsumsq, gamma, beta, scale, shift, invN);
    k_bn_relu<<<bnrelu_g, TB, 0, stream>>>(h1, scale, shift, NH);

    // ---- layer 3 (no BN/ReLU) ----
    k_fill0<<<fillNH_g, TB, 0, stream>>>(nsum, NH);
    k_aggregate<<<edgeWave_g, TB, 0, stream>>>(h1, src, dst, nsum, N_EDGES);
    k_sage_gemm<true, false, false><<<gemm_g, 128, 0, stream>>>(
        h1, nsum, deg, Wself3, Wneigh3, b3, h0, colsum, colsumsq);

    // ---- decoder MLP ----
    k_sage_gemm<false, true, false><<<gemm_g, 128, 0, stream>>>(
        h0, nullptr, nullptr, Wd1, nullptr, bd1, h1, colsum, colsumsq);
    k_sage_gemm<false, true, false><<<gemm_g, 128, 0, stream>>>(
        h1, nullptr, nullptr, Wd2, nullptr, bd2, h0, colsum, colsumsq);
    k_dot<<<dot_g, TB, 0, stream>>>(h0, Wd3, bd3, outp, N_NODES);
}